// CRF_67482526154781
// MI455X (gfx1250) — compile-verified
//
#include <hip/hip_runtime.h>
#include <hip/hip_bf16.h>

typedef __attribute__((ext_vector_type(2))) float v2f;
typedef __attribute__((ext_vector_type(8))) float v8f;

#define BDIM 256
#define TDIM 512
#define NDIM 128

// One block per batch element. 256 threads = 8 waves (wave32).
// Wave w owns output columns k = w*16 .. w*16+15.
// Lane layout matches the WMMA 16x16 C/D tiling:
//   lanes 0-15  : k = w*16 + lane,    j rows {0..7}+jt*16 in VGPRs 0..7
//   lanes 16-31 : k = w*16 + lane-16, j rows {8..15}+jt*16 in VGPRs 0..7
__global__ __launch_bounds__(256) void viterbi_kernel(
    const float* __restrict__ em,    // [B, T, N]
    const float* __restrict__ mask,  // [B, T]
    const float* __restrict__ tr,    // [N, N]
    float* __restrict__ out)         // [B] scores ++ [B, T] paths (as float)
{
    __shared__ float         s_alpha[2][NDIM];        // ping-pong
    __shared__ unsigned char s_bp[(TDIM - 1) * NDIM]; // 511*128 = 65408 B
    __shared__ unsigned char s_path[TDIM];

    const int b    = blockIdx.x;
    const int tid  = threadIdx.x;
    const int wave = tid >> 5;
    const int lane = tid & 31;
    const int l16  = lane & 15;
    const int half = lane >> 4;           // 0: j rows 0-7, 1: j rows 8-15
    const int jb   = half * 8;            // in-tile j base for this lane
    const int kcol = wave * 16 + l16;     // this lane's output column k

    // ---- Preload transitions as WMMA C-tiles into registers (64 VGPRs/wave).
    v8f ctr[8];
#pragma unroll
    for (int jt = 0; jt < 8; ++jt) {
#pragma unroll
        for (int r = 0; r < 8; ++r) {
            ctr[jt][r] = tr[(jt * 16 + jb + r) * NDIM + kcol];
        }
    }

    // ---- B operand: exactly one K-row of ones (layout-robust one-hot).
    v2f bmat;
    bmat.x = (half == 0) ? 1.0f : 0.0f;
    bmat.y = 0.0f;

    // ---- alpha0 = emissions[b, 0, :]
    const float* em_b = em + (size_t)b * TDIM * NDIM;
    if (tid < NDIM) s_alpha[0][tid] = em_b[tid];
    __syncthreads();

    // Prefetch step t=1 inputs.
    float e_cur = em_b[1 * NDIM + kcol];
    float m_cur = mask[b * TDIM + 1];

    for (int i = 0; i < TDIM - 1; ++i) {
        const int t  = i + 1;
        const int tn = (t + 1 < TDIM) ? (t + 1) : t;
        // Prefetch next step (hides global latency behind this step's math).
        float e_nxt = em_b[tn * NDIM + kcol];
        float m_nxt = mask[b * TDIM + tn];

        float* __restrict__ Acur = &s_alpha[i & 1][0];
        float* __restrict__ Anxt = &s_alpha[(i & 1) ^ 1][0];

        float aold = 0.0f;          // alpha[kcol] (old), captured at jt == wave
        float vmax = 0.0f;
        int   vidx = 0;

#pragma unroll
        for (int jt = 0; jt < 8; ++jt) {
            // A: every column = alpha[j] for this 16-row j tile (M = lane&15).
            float av = Acur[jt * 16 + l16];
            if (jt == wave) aold = av;   // Acur[kcol]
            v2f a; a.x = av; a.y = av;

            // C' = e[t,k] + tr[j,k]  (matches reference association order)
            v8f c;
#pragma unroll
            for (int r = 0; r < 8; ++r) c[r] = ctr[jt][r] + e_cur;

            // D[j,k] = alpha[j]*1 + (e+tr)[j,k]   — exact, single rounding
            v8f d = __builtin_amdgcn_wmma_f32_16x16x4_f32(
                false, a, false, bmat, (short)0, c, false, false);

            // Tournament tree (depth 3); every compare keeps the LOWER index
            // on ties (strict >), so the first max wins, like jnp.argmax.
            bool g0 = d[1] > d[0]; float w0 = g0 ? d[1] : d[0]; int p0 = g0 ? 1 : 0;
            bool g1 = d[3] > d[2]; float w1 = g1 ? d[3] : d[2]; int p1 = g1 ? 3 : 2;
            bool g2 = d[5] > d[4]; float w2 = g2 ? d[5] : d[4]; int p2 = g2 ? 5 : 4;
            bool g3 = d[7] > d[6]; float w3 = g3 ? d[7] : d[6]; int p3 = g3 ? 7 : 6;
            bool h0 = w1 > w0;     float x0 = h0 ? w1 : w0;     int q0 = h0 ? p1 : p0;
            bool h1 = w3 > w2;     float x1 = h1 ? w3 : w2;     int q1 = h1 ? p3 : p2;
            bool h2 = x1 > x0;     float wt = h2 ? x1 : x0;     int pt = h2 ? q1 : q0;
            int  jidx = jt * 16 + jb + pt;

            if (jt == 0) { vmax = wt; vidx = jidx; }
            else {
                bool gg = wt > vmax;         // earlier tile = lower j = incumbent
                vmax = gg ? wt : vmax;
                vidx = gg ? jidx : vidx;
            }
        }

        // Merge the two lane halves (same k, disjoint j sets); tie -> lower j.
        float ov = __shfl_xor(vmax, 16, 32);
        int   oi = __shfl_xor(vidx, 16, 32);
        if (ov > vmax || (ov == vmax && oi < vidx)) { vmax = ov; vidx = oi; }

        // Masked update exactly per reference.
        float na  = m_cur * vmax + (1.0f - m_cur) * aold;
        int   bpv = (m_cur != 0.0f) ? vidx : kcol;

        if (half == 0) {
            Anxt[kcol]            = na;
            s_bp[i * NDIM + kcol] = (unsigned char)bpv;
        }
        __syncthreads();   // single barrier: ping-pong makes this sufficient

        e_cur = e_nxt; m_cur = m_nxt;
    }

    // ---- Final argmax over alphas (wave 0, shfl tree), then LDS backtrace.
    const float* fin = &s_alpha[(TDIM - 1) & 1][0];
    if (tid < 32) {
        int   k0 = tid * 4;
        float bv = fin[k0];
        int   bi = k0;
#pragma unroll
        for (int r = 1; r < 4; ++r) {
            float v = fin[k0 + r];
            if (v > bv) { bv = v; bi = k0 + r; }
        }
#pragma unroll
        for (int m = 1; m < 32; m <<= 1) {
            float ov = __shfl_xor(bv, m, 32);
            int   oi = __shfl_xor(bi, m, 32);
            if (ov > bv || (ov == bv && oi < bi)) { bv = ov; bi = oi; }
        }
        if (tid == 0) {
            out[b] = bv;
            int lab = bi;
            s_path[TDIM - 1] = (unsigned char)lab;
            for (int t = TDIM - 1; t >= 1; --t) {
                lab = (int)s_bp[(t - 1) * NDIM + lab];
                s_path[t - 1] = (unsigned char)lab;
            }
        }
    }
    __syncthreads();

    // Coalesced path output (2 elements per thread).
    float* path = out + BDIM + (size_t)b * TDIM;
    path[tid]       = (float)s_path[tid];
    path[tid + 256] = (float)s_path[tid + 256];
}

extern "C" void kernel_launch(void* const* d_in, const int* in_sizes, int n_in,
                              void* d_out, int out_size, void* d_ws, size_t ws_size,
                              hipStream_t stream) {
    (void)in_sizes; (void)n_in; (void)d_ws; (void)ws_size; (void)out_size;
    const float* emissions   = (const float*)d_in[0];
    const float* mask        = (const float*)d_in[1];
    const float* transitions = (const float*)d_in[2];
    float* out = (float*)d_out;
    viterbi_kernel<<<dim3(BDIM), dim3(256), 0, stream>>>(emissions, mask,
                                                         transitions, out);
}